// RecyclingEmbedder_30090540875865
// MI455X (gfx1250) — compile-verified
//
#include <hip/hip_runtime.h>
#include <hip/hip_bf16.h>
#include <stdint.h>

// ---------------------------------------------------------------------------
// RecyclingEmbedder fused kernels for MI455X (gfx1250, wave32)
//
// Roofline: z stream = 512 MB read + 512 MB write -> ~45 us @ 23.3 TB/s.
// Everything (distogram one-hot linear via V_WMMA_F32_16X16X4_F32, LayerNorm,
// bias adds) is fused into a single pass over z. z tiles are staged
// global->LDS with CDNA5 async-to-LDS transfers (ASYNCcnt).
// ---------------------------------------------------------------------------

#define N_RES    1024
#define C_M      256
#define C_Z      128
#define NUM_BINS 15
#define ZSTRIDE  132          // padded LDS row stride (floats); 132*4 % 16 == 0
#define ZWAVES   4            // waves (16-pair tiles) per block

typedef __attribute__((ext_vector_type(2))) float v2f;
typedef __attribute__((ext_vector_type(8))) float v8f;

// ---------------------------------------------------------------------------
// Kernel 1: LayerNorm of m  [1024 x 256]  (one wave32 per row)
// ---------------------------------------------------------------------------
__global__ __launch_bounds__(128)
void m_ln_kernel(const float* __restrict__ m,
                 const float* __restrict__ w,
                 const float* __restrict__ b,
                 float* __restrict__ out)
{
    const int lane = threadIdx.x & 31;
    const int wave = threadIdx.x >> 5;
    const int row  = blockIdx.x * ZWAVES + wave;
    if (row >= N_RES) return;

    const float* src = m + (size_t)row * C_M;
    float4 v0 = *(const float4*)(src + 4 * lane);          // channels 4l..4l+3
    float4 v1 = *(const float4*)(src + 128 + 4 * lane);    // channels 128+4l..

    float s  = v0.x + v0.y + v0.z + v0.w + v1.x + v1.y + v1.z + v1.w;
    float ss = v0.x*v0.x + v0.y*v0.y + v0.z*v0.z + v0.w*v0.w
             + v1.x*v1.x + v1.y*v1.y + v1.z*v1.z + v1.w*v1.w;
    #pragma unroll
    for (int o = 16; o >= 1; o >>= 1) {
        s  += __shfl_xor(s,  o, 32);
        ss += __shfl_xor(ss, o, 32);
    }
    const float mu  = s * (1.0f / C_M);
    const float var = ss * (1.0f / C_M) - mu * mu;
    const float rs  = rsqrtf(var + 1e-5f);

    float4 w0 = *(const float4*)(w + 4 * lane);
    float4 w1 = *(const float4*)(w + 128 + 4 * lane);
    float4 b0 = *(const float4*)(b + 4 * lane);
    float4 b1 = *(const float4*)(b + 128 + 4 * lane);

    float4 o0, o1;
    o0.x = (v0.x - mu) * rs * w0.x + b0.x;  o0.y = (v0.y - mu) * rs * w0.y + b0.y;
    o0.z = (v0.z - mu) * rs * w0.z + b0.z;  o0.w = (v0.w - mu) * rs * w0.w + b0.w;
    o1.x = (v1.x - mu) * rs * w1.x + b1.x;  o1.y = (v1.y - mu) * rs * w1.y + b1.y;
    o1.z = (v1.z - mu) * rs * w1.z + b1.z;  o1.w = (v1.w - mu) * rs * w1.w + b1.w;

    float* dst = out + (size_t)row * C_M;
    *(float4*)(dst + 4 * lane)       = o0;
    *(float4*)(dst + 128 + 4 * lane) = o1;
}

// ---------------------------------------------------------------------------
// Kernel 2: fused  z_update = LN(z) + onehot(d2) @ lin_w^T + lin_b
// One wave32 per 16-pair tile (16 consecutive j for one i; rows contiguous).
// ---------------------------------------------------------------------------
__global__ __launch_bounds__(32 * ZWAVES)
void z_fused_kernel(const float* __restrict__ z,
                    const float* __restrict__ x,
                    const float* __restrict__ ln_z_w,
                    const float* __restrict__ ln_z_b,
                    const float* __restrict__ lin_w,
                    const float* __restrict__ lin_b,
                    float* __restrict__ zout)
{
    __shared__ __align__(16) float wlds[16 * C_Z];                    // lin_w^T, bin-major, bin 15 = 0
    __shared__ __align__(16) float zlds[ZWAVES * 16 * ZSTRIDE];       // staged z tiles (padded rows)

    const int tid  = threadIdx.x;
    const int lane = tid & 31;
    const int wave = tid >> 5;
    const int p16  = lane & 15;            // pair index within tile (= WMMA N / A-row M)
    const int hi   = (lane >> 4) & 1;      // half-wave selector

    // ---- stage lin_w transposed into LDS (block-cooperative) ----
    for (int idx = tid; idx < 16 * C_Z; idx += blockDim.x) {
        const int bin = idx >> 7;          // / 128
        const int c   = idx & 127;
        wlds[idx] = (bin < NUM_BINS) ? lin_w[c * NUM_BINS + bin] : 0.0f;
    }

    // ---- async-stage this wave's 16 z rows (8 KB contiguous) into LDS ----
    const int tile = blockIdx.x * ZWAVES + wave;       // 65536 tiles total
    const int i    = tile >> 6;                        // 64 tiles per i-row
    const int jb   = (tile & 63) << 4;
    const long long pair_base = (long long)i * N_RES + jb;
    const float* zsrc   = z + pair_base * C_Z;
    float*       my_lds = &zlds[wave * 16 * ZSTRIDE];

    #pragma unroll
    for (int p = 0; p < 16; ++p) {
        // 32 lanes x 16B = one full 512B row per instruction
        uint32_t ldsoff = (uint32_t)(uintptr_t)(&my_lds[p * ZSTRIDE]) + (uint32_t)(lane * 16);
        uint64_t gaddr  = (uint64_t)(uintptr_t)(zsrc + p * C_Z) + (uint64_t)(lane * 16);
        asm volatile("global_load_async_to_lds_b128 %0, %1, off"
                     :: "v"(ldsoff), "v"(gaddr) : "memory");
    }

    __syncthreads();   // wlds visible to all waves (z tiles are wave-private)

    // ---- distance bin for this lane's pair (exact f32, matches reference) ----
    const int jj = jb + p16;
    const float dx = x[i * 3 + 0] - x[jj * 3 + 0];
    const float dy = x[i * 3 + 1] - x[jj * 3 + 1];
    const float dz = x[i * 3 + 2] - x[jj * 3 + 2];
    const float d2 = dx * dx + dy * dy + dz * dz;
    int bin = -1;
    #pragma unroll
    for (int k = 0; k < NUM_BINS; ++k) {
        float lo = 3.25f + 1.25f * (float)k;       lo *= lo;
        float up;
        if (k < NUM_BINS - 1) { up = 3.25f + 1.25f * (float)(k + 1); up *= up; }
        else                  { up = 1e8f; }
        if (d2 > lo && d2 < up) bin = k;
    }

    // ---- one-hot A matrix in WMMA 16x4xF32 layout, 4 K-chunks of 4 bins ----
    // A layout: VGPR0 = K (0 | 2) per half-wave, VGPR1 = K (1 | 3); M = p16.
    v2f A[4];
    #pragma unroll
    for (int k = 0; k < 4; ++k) {
        const int k0 = 4 * k + 2 * hi;
        A[k].x = (bin == k0)     ? 1.0f : 0.0f;
        A[k].y = (bin == k0 + 1) ? 1.0f : 0.0f;
    }

    // ---- wait for async staging, compute per-pair LN stats from LDS ----
    asm volatile("s_wait_asynccnt 0x0" ::: "memory");

    float s = 0.0f, ss = 0.0f;
    {
        const float* row = &my_lds[p16 * ZSTRIDE + hi * 64];   // half the channels each
        #pragma unroll
        for (int q = 0; q < 16; ++q) {
            float4 v = *(const float4*)(row + 4 * q);
            s  += v.x + v.y + v.z + v.w;
            ss += v.x * v.x + v.y * v.y + v.z * v.z + v.w * v.w;
        }
    }
    s  += __shfl_xor(s,  16, 32);
    ss += __shfl_xor(ss, 16, 32);
    const float mu  = s * (1.0f / C_Z);
    const float var = ss * (1.0f / C_Z) - mu * mu;
    const float rs  = rsqrtf(var + 1e-5f);

    // broadcast stats for the 8 pairs this lane touches in the C/D layout
    float mu8[8], rs8[8];
    #pragma unroll
    for (int r = 0; r < 8; ++r) {
        const int p = r + 8 * hi;
        mu8[r] = __shfl(mu, p, 32);
        rs8[r] = __shfl(rs, p, 32);
    }

    float* my_out = zout + pair_base * C_Z;

    // ---- 8 channel tiles: D = onehot x lin_w^T via chained f32 WMMA ----
    #pragma unroll
    for (int t = 0; t < 8; ++t) {
        const int c = 16 * t + p16;        // channel handled by this lane (N = p16)

        v8f acc = {0.f, 0.f, 0.f, 0.f, 0.f, 0.f, 0.f, 0.f};
        #pragma unroll
        for (int k = 0; k < 4; ++k) {
            // B layout (4x16): VGPR0 = row K (0 | 2), VGPR1 = row K (1 | 3)
            v2f B;
            B.x = wlds[(4 * k + 2 * hi    ) * C_Z + c];
            B.y = wlds[(4 * k + 2 * hi + 1) * C_Z + c];
            acc = __builtin_amdgcn_wmma_f32_16x16x4_f32(
                      /*neg_a=*/false, A[k], /*neg_b=*/false, B,
                      /*c_mod=*/(short)0, acc, /*reuse_a=*/false, /*reuse_b=*/false);
        }

        const float lw = ln_z_w[c];
        const float lb = ln_z_b[c];
        const float ob = lin_b[c];

        // C/D layout: VGPR r -> pair (r + 8*hi), column p16
        #pragma unroll
        for (int r = 0; r < 8; ++r) {
            const int p = r + 8 * hi;
            const float zv = my_lds[p * ZSTRIDE + c];
            my_out[p * C_Z + c] = (zv - mu8[r]) * rs8[r] * lw + lb + acc[r] + ob;
        }
    }
}

// ---------------------------------------------------------------------------
// Launch
// ---------------------------------------------------------------------------
extern "C" void kernel_launch(void* const* d_in, const int* in_sizes, int n_in,
                              void* d_out, int out_size, void* d_ws, size_t ws_size,
                              hipStream_t stream)
{
    (void)in_sizes; (void)n_in; (void)out_size; (void)d_ws; (void)ws_size;

    const float* m      = (const float*)d_in[0];
    const float* z      = (const float*)d_in[1];
    const float* x      = (const float*)d_in[2];
    const float* ln_m_w = (const float*)d_in[3];
    const float* ln_m_b = (const float*)d_in[4];
    const float* ln_z_w = (const float*)d_in[5];
    const float* ln_z_b = (const float*)d_in[6];
    const float* lin_w  = (const float*)d_in[7];
    const float* lin_b  = (const float*)d_in[8];

    float* m_out = (float*)d_out;                      // [1024*256]
    float* z_out = m_out + (size_t)N_RES * C_M;        // [1024*1024*128]

    m_ln_kernel<<<N_RES / ZWAVES, 32 * ZWAVES, 0, stream>>>(m, ln_m_w, ln_m_b, m_out);

    const int tiles  = (N_RES * N_RES) / 16;           // 65536 16-pair tiles
    const int blocks = tiles / ZWAVES;                 // 16384 blocks of 4 waves
    z_fused_kernel<<<blocks, 32 * ZWAVES, 0, stream>>>(z, x, ln_z_w, ln_z_b,
                                                       lin_w, lin_b, z_out);
}